// MaxPoolBlock_15942918603361
// MI455X (gfx1250) — compile-verified
//
#include <hip/hip_runtime.h>
#include <math.h>

// Gather + row-max pooling, tuned for MI455X (gfx1250, wave32).
//
// out[r, :] = max_{j in 0..31} ( pools[r,j] == n1 ? 0 : x[pools[r,j], :] )
//
// x (102.4 MB) is L2-resident on MI455X (192 MB L2); the kernel is bound by
// L2 gather bandwidth/latency. Mapping: one wave32 per output row; lane l
// owns columns [4l, 4l+4) as a float4, so each gathered row is one coalesced
// 512B global_load_b128 per wave (saddr form via v_readlane-broadcast
// indices). The gather loop is software-pipelined in batches of 8 with
// double buffering so 8 b128 loads are always in flight per wave under the
// gfx1250 split s_wait_loadcnt counters. Shadow index (== n1) is handled
// branchlessly with v_cndmask. global_prefetch_b8 bulk-prefetches the wave's
// whole 16KB gather set up front.

constexpr int D               = 128;  // feature dim (reference)
constexpr int MAXN            = 32;   // pool size (reference)
constexpr int WAVES_PER_BLOCK = 8;    // 256 threads = 8 wave32
constexpr int THREADS         = WAVES_PER_BLOCK * 32;
constexpr int BATCH           = 8;    // gathers in flight per wave
constexpr int NBATCH          = MAXN / BATCH;

__global__ __launch_bounds__(THREADS)
void maxpool_gather_kernel(const float* __restrict__ x,
                           const int*   __restrict__ pools,
                           float*       __restrict__ out,
                           int n1, int n2)
{
    const int lane = threadIdx.x & 31;
    const int wave = threadIdx.x >> 5;
    const int row  = blockIdx.x * WAVES_PER_BLOCK + wave;
    if (row >= n2) return;

    // One coalesced b32 load: lane l holds pool index j=l for this row.
    const int myidx = pools[(size_t)row * MAXN + lane];

    // Bulk prefetch: lane l prefetches all four 128B lines of row pools[r,l].
    // One wave-wide burst covers the wave's entire 32-row x 512B gather set.
    {
        const int safe_pf = (myidx < n1) ? myidx : 0;
        const float* rbase = x + (size_t)safe_pf * D;
        __builtin_prefetch(rbase +  0, 0, 3);   // global_prefetch_b8
        __builtin_prefetch(rbase + 32, 0, 3);
        __builtin_prefetch(rbase + 64, 0, 3);
        __builtin_prefetch(rbase + 96, 0, 3);
    }

    // Broadcast all 32 indices to SGPRs (v_readlane, immediate lane).
    int idx[MAXN];
#pragma unroll
    for (int j = 0; j < MAXN; ++j)
        idx[j] = __builtin_amdgcn_readlane(myidx, j);

    const int col = lane << 2;  // 4 floats per lane
    float4 acc = make_float4(-INFINITY, -INFINITY, -INFINITY, -INFINITY);

    // Software-pipelined gather: double-buffered batches of 8 b128 loads.
    float4 buf[BATCH];
#pragma unroll
    for (int j = 0; j < BATCH; ++j) {
        const int safe = (idx[j] < n1) ? idx[j] : 0;
        buf[j] = *reinterpret_cast<const float4*>(x + (size_t)safe * D + col);
    }

#pragma unroll
    for (int b = 0; b < NBATCH; ++b) {
        float4 cur[BATCH];
#pragma unroll
        for (int j = 0; j < BATCH; ++j) cur[j] = buf[j];

        // Issue next batch before consuming the current one: keeps 8 loads
        // outstanding while the VALU reduce runs.
        if (b + 1 < NBATCH) {
#pragma unroll
            for (int j = 0; j < BATCH; ++j) {
                const int jj   = (b + 1) * BATCH + j;
                const int safe = (idx[jj] < n1) ? idx[jj] : 0;
                buf[j] = *reinterpret_cast<const float4*>(
                    x + (size_t)safe * D + col);
            }
        }

#pragma unroll
        for (int j = 0; j < BATCH; ++j) {
            // Branchless shadow-row handling: invalid index contributes 0.0
            // (matches the reference's appended zero row).
            const bool ok = (idx[b * BATCH + j] < n1);
            const float4 v = cur[j];
            acc.x = fmaxf(acc.x, ok ? v.x : 0.0f);
            acc.y = fmaxf(acc.y, ok ? v.y : 0.0f);
            acc.z = fmaxf(acc.z, ok ? v.z : 0.0f);
            acc.w = fmaxf(acc.w, ok ? v.w : 0.0f);
        }
    }

    *reinterpret_cast<float4*>(out + (size_t)row * D + col) = acc;
}

extern "C" void kernel_launch(void* const* d_in, const int* in_sizes, int n_in,
                              void* d_out, int out_size, void* d_ws, size_t ws_size,
                              hipStream_t stream)
{
    const float* x     = (const float*)d_in[0];
    const int*   pools = (const int*)d_in[1];
    float*       out   = (float*)d_out;

    const int n1 = in_sizes[0] / D;     // 200000 rows in x
    const int n2 = in_sizes[1] / MAXN;  // 50000 output rows

    const int blocks = (n2 + WAVES_PER_BLOCK - 1) / WAVES_PER_BLOCK;
    maxpool_gather_kernel<<<dim3(blocks), dim3(THREADS), 0, stream>>>(
        x, pools, out, n1, n2);
}